// GINModel_1486058684701
// MI455X (gfx1250) — compile-verified
//
#include <hip/hip_runtime.h>

#define N_NODES 100000
#define HIDDEN  64
#define NLAYER  3
#define OUTD    10
#define CONCATD (HIDDEN * (NLAYER + 1))   // 256
#define BN_EPS  1e-5f

typedef __attribute__((ext_vector_type(2))) float v2f;
typedef __attribute__((ext_vector_type(8))) float v8f;

// ---- gfx1250 async global->LDS copy (ASYNCcnt path), with safe fallback ----
#if defined(__has_builtin)
#if __has_builtin(__builtin_amdgcn_global_load_async_to_lds_b128) && \
    __has_builtin(__builtin_amdgcn_s_wait_asynccnt)
#define HAVE_ASYNC_LDS 1
#endif
#endif

#ifdef HAVE_ASYNC_LDS
// diagnostic from this toolchain: param is "int __vector(4) __device__ *",
// i.e. AS1 (global) pointer to a 16-byte int vector; LDS side is AS3.
typedef int v4i __attribute__((vector_size(16)));
typedef __attribute__((address_space(1))) v4i as1_v4i;
typedef __attribute__((address_space(3))) v4i as3_v4i;

__device__ __forceinline__ void async_copy_b128(void* lds, const void* gptr)
{
    // generic LDS pointers keep the LDS byte offset in the low 32 bits;
    // generic global pointers are value-identical to AS1 pointers.
    __builtin_amdgcn_global_load_async_to_lds_b128(
        (as1_v4i*)(unsigned long long)(uintptr_t)gptr,
        (as3_v4i*)(unsigned)(uintptr_t)lds,
        0, 0);
}
#endif

// ---------------------------------------------------------------------------
// GEMM  C(N x 64) = A(N x K, row stride lda) * B(K x 64) + bias, fp32 WMMA.
// Also accumulates per-column sum / sum-of-squares for BatchNorm stats.
// 4 waves per block; each wave owns one 16-col tile and TWO 16-row tiles
// (independent accumulator chains; B fragments shared between them).
// B is staged through LDS in 64-row chunks via async-to-LDS when available.
// ---------------------------------------------------------------------------
__global__ __launch_bounds__(128)
void gemm_bn_stats(const float* __restrict__ A, int lda,
                   const float* __restrict__ B,       // K x 64 row-major
                   const float* __restrict__ bias,    // 64
                   float* __restrict__ C,             // N x 64 dense
                   float* __restrict__ gsum, float* __restrict__ gsumsq,
                   int K)
{
    __shared__ float Bs[64 * 64];     // one K-chunk of B (16 KB)
    __shared__ float ls[64], lq[64];

    const int tid  = threadIdx.x;
    const int wave = tid >> 5;
    const int lane = tid & 31;
    const int n    = lane & 15;       // row-in-tile (A) / col-in-tile (B,C)
    const int half = lane >> 4;       // K-half selector per ISA layout
    const int col  = wave * 16 + n;   // output column 0..63
    const long r0  = (long)blockIdx.x * 32 + n;   // row-tile 0
    const long r1  = r0 + 16;                     // row-tile 1

    v8f acc0 = {}, acc1 = {};
    for (int kb = 0; kb < K; kb += 64) {
        // stage B rows [kb, kb+64) into LDS (contiguous 4096 floats)
        const float* bsrc = B + (size_t)kb * 64;
#ifdef HAVE_ASYNC_LDS
        for (int i = tid; i < (64 * 64) / 4; i += 128)
            async_copy_b128(&Bs[i * 4], bsrc + (size_t)i * 4);
        __builtin_amdgcn_s_wait_asynccnt(0);
#else
        for (int i = tid; i < (64 * 64) / 4; i += 128)
            ((float4*)Bs)[i] = ((const float4*)bsrc)[i];
#endif
        __syncthreads();

#pragma unroll
        for (int kk = 0; kk < 16; ++kk) {
            // A 16x4 fragment: lanes 0-15 hold K = {0,1}, lanes 16-31 hold K = {2,3}
            const int  kl = kk * 4 + half * 2;      // within chunk
            const long ak = (long)kb + kl;
            v2f a0, a1, b;
            a0.x = A[r0 * lda + ak];
            a0.y = A[r0 * lda + ak + 1];
            a1.x = A[r1 * lda + ak];
            a1.y = A[r1 * lda + ak + 1];
            // B 4x16 fragment: VGPR v at lane-half h holds row K = 2h + v
            b.x = Bs[kl * 64 + col];
            b.y = Bs[(kl + 1) * 64 + col];
            acc0 = __builtin_amdgcn_wmma_f32_16x16x4_f32(
                       false, a0, false, b, (short)0, acc0, false, false);
            acc1 = __builtin_amdgcn_wmma_f32_16x16x4_f32(
                       false, a1, false, b, (short)0, acc1, false, false);
        }
        __syncthreads();
    }

    // bias + store + local BN-stat accumulation
    const float bv = bias[col];
    float s = 0.f, q = 0.f;
#pragma unroll
    for (int r = 0; r < 8; ++r) {
        const float v0 = acc0[r] + bv;
        const float v1 = acc1[r] + bv;
        s += v0 + v1;
        q += v0 * v0 + v1 * v1;
        const long ro = (long)blockIdx.x * 32 + r + half * 8;  // C row per ISA C-layout
        C[ro * 64 + col]        = v0;
        C[(ro + 16) * 64 + col] = v1;
    }

    if (tid < 64) { ls[tid] = 0.f; lq[tid] = 0.f; }
    __syncthreads();
    unsafeAtomicAdd(&ls[col], s);
    unsafeAtomicAdd(&lq[col], q);
    __syncthreads();
    if (tid < 64) {
        unsafeAtomicAdd(&gsum[tid],   ls[tid]);
        unsafeAtomicAdd(&gsumsq[tid], lq[tid]);
    }
}

// zero the 128-float stats region
__global__ void zero_stats(float* p)
{
    if (threadIdx.x < 128) p[threadIdx.x] = 0.f;
}

// fold BN into per-column scale/shift: y' = y*scale + shift
__global__ void bn_finalize(const float* __restrict__ gsum,
                            const float* __restrict__ gsumsq,
                            const float* __restrict__ g,
                            const float* __restrict__ be,
                            float* __restrict__ scale, float* __restrict__ shift,
                            float invN)
{
    const int c = threadIdx.x;   // blockDim == 64
    const float mean = gsum[c] * invN;
    const float var  = gsumsq[c] * invN - mean * mean;   // biased variance
    const float inv  = rsqrtf(var + BN_EPS);
    const float sc   = g[c] * inv;
    scale[c] = sc;
    shift[c] = be[c] - mean * sc;
}

// out[n*ostride + c] = relu(in[n*64 + c] * scale[c] + shift[c])
__global__ __launch_bounds__(256)
void bn_apply_relu(const float* __restrict__ in, float* __restrict__ out,
                   int ostride, const float* __restrict__ scale,
                   const float* __restrict__ shift)
{
    const long i = (long)blockIdx.x * 256 + threadIdx.x;
    if (i >= (long)N_NODES * 64) return;
    const int  c = (int)(i & 63);
    const long nrow = i >> 6;
    const float v = in[i] * scale[c] + shift[c];
    out[nrow * ostride + c] = fmaxf(v, 0.f);
}

// aggr[n][c] = (1+eps[li]) * h[n][c]   (h read strided out of concat buffer)
__global__ __launch_bounds__(256)
void init_aggr(const float* __restrict__ hcat, int coff,
               const float* __restrict__ eps, int li,
               float* __restrict__ aggr)
{
    const long i = (long)blockIdx.x * 256 + threadIdx.x;
    if (i >= (long)N_NODES * 64) return;
    const int  c = (int)(i & 63);
    const long nrow = i >> 6;
    aggr[i] = (1.0f + eps[li]) * hcat[nrow * CONCATD + coff + c];
}

// segment_sum: aggr[dst[e]] += h[src[e]]; thread = (edge, 4-feature chunk)
__global__ __launch_bounds__(256)
void scatter_add(const int* __restrict__ src, const int* __restrict__ dst,
                 long E, const float* __restrict__ hcat, int coff,
                 float* __restrict__ aggr)
{
    const long t = (long)blockIdx.x * 256 + threadIdx.x;
    if (t >= E * 16) return;
    const long e = t >> 4;
    const int  q = (int)(t & 15);
    const int  s = src[e];
    const int  d = dst[e];
    const float4 v = *(const float4*)(hcat + (long)s * CONCATD + coff + q * 4);
    float* p = aggr + (long)d * 64 + q * 4;
    unsafeAtomicAdd(p + 0, v.x);
    unsafeAtomicAdd(p + 1, v.y);
    unsafeAtomicAdd(p + 2, v.z);
    unsafeAtomicAdd(p + 3, v.w);
}

// column sums / sums-of-squares over a dense N x 64 buffer (for the post-BN)
__global__ __launch_bounds__(256)
void col_stats(const float* __restrict__ in, float* __restrict__ gsum,
               float* __restrict__ gsumsq)
{
    __shared__ float ls[64], lq[64];
    const int tid = threadIdx.x;
    if (tid < 64) { ls[tid] = 0.f; lq[tid] = 0.f; }
    __syncthreads();
    const int  c  = tid & 63;
    const long r0 = (long)blockIdx.x * 4 + (tid >> 6);
    const long st = (long)gridDim.x * 4;
    float s = 0.f, q = 0.f;
    for (long r = r0; r < N_NODES; r += st) {
        const float v = in[r * 64 + c];
        s += v;
        q += v * v;
    }
    unsafeAtomicAdd(&ls[c], s);
    unsafeAtomicAdd(&lq[c], q);
    __syncthreads();
    if (tid < 64) {
        unsafeAtomicAdd(&gsum[tid],   ls[tid]);
        unsafeAtomicAdd(&gsumsq[tid], lq[tid]);
    }
}

// final classifier: out(N x 10) = A(N x 64) @ W(64 x 10) + b   (tiny, scalar)
__global__ __launch_bounds__(256)
void out_gemm(const float* __restrict__ A, const float* __restrict__ W,
              const float* __restrict__ b, float* __restrict__ out)
{
    const long t = (long)blockIdx.x * 256 + threadIdx.x;
    if (t >= (long)N_NODES * 16) return;
    const long nrow = t >> 4;
    const int  o = (int)(t & 15);
    if (o >= OUTD) return;
    float acc = b[o];
    const float* a = A + nrow * 64;
#pragma unroll
    for (int k = 0; k < 64; ++k) acc += a[k] * W[k * OUTD + o];
    out[nrow * OUTD + o] = acc;
}

// ---------------------------------------------------------------------------
extern "C" void kernel_launch(void* const* d_in, const int* in_sizes, int n_in,
                              void* d_out, int out_size, void* d_ws, size_t ws_size,
                              hipStream_t stream)
{
    const float* x       = (const float*)d_in[0];
    const int*   ei      = (const int*)  d_in[1];
    const float* W_in    = (const float*)d_in[2];
    const float* b_in    = (const float*)d_in[3];
    const float* g_in    = (const float*)d_in[4];
    const float* be_in   = (const float*)d_in[5];
    const float* eps     = (const float*)d_in[6];
    const float* W1      = (const float*)d_in[7];
    const float* b1      = (const float*)d_in[8];
    const float* g1      = (const float*)d_in[9];
    const float* be1     = (const float*)d_in[10];
    const float* W2      = (const float*)d_in[11];
    const float* b2      = (const float*)d_in[12];
    const float* g2      = (const float*)d_in[13];
    const float* be2     = (const float*)d_in[14];
    const float* g_post  = (const float*)d_in[15];
    const float* be_post = (const float*)d_in[16];
    const float* W_out1  = (const float*)d_in[17];
    const float* b_out1  = (const float*)d_in[18];
    const float* g_out   = (const float*)d_in[19];
    const float* be_out  = (const float*)d_in[20];
    const float* W_out2  = (const float*)d_in[21];
    const float* b_out2  = (const float*)d_in[22];

    const long E = (long)in_sizes[1] / 2;
    const int* src = ei;
    const int* dst = ei + E;

    // workspace layout
    float* concat = (float*)d_ws;                                  // N x 256
    float* bufA   = concat + (size_t)N_NODES * CONCATD;            // N x 64
    float* bufB   = bufA   + (size_t)N_NODES * HIDDEN;             // N x 64
    float* st     = bufB   + (size_t)N_NODES * HIDDEN;             // 256 floats
    float* gsum = st, *gsq = st + 64, *sc = st + 128, *sh = st + 192;

    const float invN = 1.0f / (float)N_NODES;
    const int gemm_blocks  = N_NODES / 32;                         // 3125
    const int elem_blocks  = (N_NODES * 64) / 256;                 // 25000
    const int edge_blocks  = (int)((E * 16 + 255) / 256);          // 100000
    const int ogemm_blocks = (N_NODES * 16) / 256;                 // 6250

    // ---- input projection: h0 = relu(BN(x @ W_in + b_in)) -> concat cols 0..63
    zero_stats<<<1, 128, 0, stream>>>(gsum);
    gemm_bn_stats<<<gemm_blocks, 128, 0, stream>>>(x, HIDDEN, W_in, b_in,
                                                   bufA, gsum, gsq, HIDDEN);
    bn_finalize<<<1, 64, 0, stream>>>(gsum, gsq, g_in, be_in, sc, sh, invN);
    bn_apply_relu<<<elem_blocks, 256, 0, stream>>>(bufA, concat, CONCATD, sc, sh);

    // ---- GIN layers
    for (int i = 0; i < NLAYER; ++i) {
        const int coff = i * HIDDEN;
        // aggr = (1+eps)*h ; then scatter-add neighbors
        init_aggr<<<elem_blocks, 256, 0, stream>>>(concat, coff, eps, i, bufB);
        scatter_add<<<edge_blocks, 256, 0, stream>>>(src, dst, E, concat, coff, bufB);

        // MLP layer 1
        zero_stats<<<1, 128, 0, stream>>>(gsum);
        gemm_bn_stats<<<gemm_blocks, 128, 0, stream>>>(bufB, HIDDEN,
                W1 + (size_t)i * HIDDEN * HIDDEN, b1 + i * HIDDEN,
                bufA, gsum, gsq, HIDDEN);
        bn_finalize<<<1, 64, 0, stream>>>(gsum, gsq, g1 + i * HIDDEN,
                                          be1 + i * HIDDEN, sc, sh, invN);
        bn_apply_relu<<<elem_blocks, 256, 0, stream>>>(bufA, bufB, HIDDEN, sc, sh);

        // MLP layer 2
        zero_stats<<<1, 128, 0, stream>>>(gsum);
        gemm_bn_stats<<<gemm_blocks, 128, 0, stream>>>(bufB, HIDDEN,
                W2 + (size_t)i * HIDDEN * HIDDEN, b2 + i * HIDDEN,
                bufA, gsum, gsq, HIDDEN);
        bn_finalize<<<1, 64, 0, stream>>>(gsum, gsq, g2 + i * HIDDEN,
                                          be2 + i * HIDDEN, sc, sh, invN);
        bn_apply_relu<<<elem_blocks, 256, 0, stream>>>(bufA, bufB, HIDDEN, sc, sh);

        // post BN (no matmul) -> h_{i+1} into concat cols [64*(i+1), ...)
        zero_stats<<<1, 128, 0, stream>>>(gsum);
        col_stats<<<1024, 256, 0, stream>>>(bufB, gsum, gsq);
        bn_finalize<<<1, 64, 0, stream>>>(gsum, gsq, g_post + i * HIDDEN,
                                          be_post + i * HIDDEN, sc, sh, invN);
        bn_apply_relu<<<elem_blocks, 256, 0, stream>>>(bufB,
                concat + (i + 1) * HIDDEN, CONCATD, sc, sh);
    }

    // ---- output head: relu(BN(concat @ W_out1 + b_out1)) @ W_out2 + b_out2
    zero_stats<<<1, 128, 0, stream>>>(gsum);
    gemm_bn_stats<<<gemm_blocks, 128, 0, stream>>>(concat, CONCATD, W_out1,
                                                   b_out1, bufA, gsum, gsq, CONCATD);
    bn_finalize<<<1, 64, 0, stream>>>(gsum, gsq, g_out, be_out, sc, sh, invN);
    bn_apply_relu<<<elem_blocks, 256, 0, stream>>>(bufA, bufB, HIDDEN, sc, sh);

    out_gemm<<<ogemm_blocks, 256, 0, stream>>>(bufB, W_out2, b_out2, (float*)d_out);
}